// MinimalController_77068893160303
// MI455X (gfx1250) — compile-verified
//
#include <hip/hip_runtime.h>
#include <math.h>

#define B_ROWS 2048
#define S_LEN 2048
#define HID 64
#define MSLOTS 8
#define VOCAB 128

typedef __attribute__((ext_vector_type(2))) float v2f;
typedef __attribute__((ext_vector_type(8))) float v8f;

// ---------------------------------------------------------------------------
// Kernel 1: per-token tables (128 tokens only).
//   token_sig[t]   = sigmoid(embed[t] . wgate_w + wgate_b)   (write prob table)
//   premem[t]      = embed[t] . rgate_w[64:128]              (read-gate mem part)
//   rank_of[t]     = rank of token by descending write score (stable by token id)
//   token_of_rank[r] = inverse permutation
// ---------------------------------------------------------------------------
__global__ void token_tables_kernel(const float* __restrict__ embed,
                                    const float* __restrict__ wgate_w,
                                    const float* __restrict__ wgate_b,
                                    const float* __restrict__ rgate_w,
                                    float* __restrict__ token_sig,
                                    float* __restrict__ premem,
                                    int* __restrict__ rank_of,
                                    int* __restrict__ token_of_rank) {
    __shared__ float sc[VOCAB];
    int t = threadIdx.x;  // 128 threads
    float d0 = 0.f, d1 = 0.f;
    for (int j = 0; j < HID; ++j) {
        float e = embed[t * HID + j];
        d0 += e * wgate_w[j];
        d1 += e * rgate_w[HID + j];
    }
    float lg = d0 + wgate_b[0];
    sc[t] = lg;
    token_sig[t] = 1.f / (1.f + expf(-lg));
    premem[t] = d1;
    __syncthreads();
    int r = 0;
    for (int u = 0; u < VOCAB; ++u) {
        float su = sc[u];
        if (su > lg || (su == lg && u < t)) r++;
    }
    rank_of[t] = r;
    token_of_rank[r] = t;
}

// ---------------------------------------------------------------------------
// Kernel 2: one block (256 threads) per batch row. Streams seq once:
//   - writes write_prob
//   - builds 18-bit keys (rank<<11 | s) in LDS, extracts 8 smallest (top-k)
//   - token histogram -> ctx, read gate, retrieved vector
// ---------------------------------------------------------------------------
__global__ void __launch_bounds__(256)
row_kernel(const int* __restrict__ seq,
           const float* __restrict__ embed,
           const float* __restrict__ rgate_w,
           const float* __restrict__ rgate_b,
           const float* __restrict__ token_sig,
           const float* __restrict__ premem,
           const int* __restrict__ rank_of,
           const int* __restrict__ token_of_rank,
           float* __restrict__ write_prob_out,   // (B,S)
           float* __restrict__ read_prob_out,    // (B,M)
           float* __restrict__ retrieved_out) {  // (B,H) scratch
    __shared__ unsigned keys[S_LEN];       // 8 KB
    __shared__ unsigned red[256];
    __shared__ unsigned cnt[VOCAB];
    __shared__ float ctxv[HID];
    __shared__ unsigned selkey[MSLOTS];
    __shared__ float rp[MSLOTS];
    __shared__ int seltok[MSLOTS];
    __shared__ float ctxdot;

    const int b = blockIdx.x;
    const int tid = threadIdx.x;
    const int* srow = seq + (size_t)b * S_LEN;

    __builtin_prefetch(srow + tid * 8, 0, 0);  // global_prefetch_b8

    if (tid < VOCAB) cnt[tid] = 0u;
    __syncthreads();

    // Stream the row: write_prob, keys, counts.
    #pragma unroll
    for (int i = 0; i < S_LEN / 256; ++i) {
        int s = tid + i * 256;
        int tok = srow[s];
        write_prob_out[(size_t)b * S_LEN + s] = token_sig[tok];
        keys[s] = ((unsigned)rank_of[tok] << 11) | (unsigned)s;
        atomicAdd(&cnt[tok], 1u);
    }
    __syncthreads();

    // Extract the 8 smallest keys (== top-8 by score desc, index asc).
    for (int m = 0; m < MSLOTS; ++m) {
        unsigned mn = 0xFFFFFFFFu;
        #pragma unroll
        for (int i = 0; i < S_LEN / 256; ++i) {
            unsigned k = keys[tid + i * 256];
            mn = (k < mn) ? k : mn;
        }
        red[tid] = mn;
        __syncthreads();
        for (int off = 128; off > 0; off >>= 1) {
            if (tid < off) {
                unsigned o = red[tid + off];
                if (o < red[tid]) red[tid] = o;
            }
            __syncthreads();
        }
        if (tid == 0) {
            unsigned k = red[0];
            selkey[m] = k;
            keys[k & (S_LEN - 1)] = 0xFFFFFFFFu;  // remove from pool
        }
        __syncthreads();
    }

    // ctx = histogram-weighted mean embedding.
    if (tid < HID) {
        float s = 0.f;
        for (int t = 0; t < VOCAB; ++t) s += (float)cnt[t] * embed[t * HID + tid];
        ctxv[tid] = s * (1.f / (float)S_LEN);
    }
    __syncthreads();
    if (tid == 0) {
        float d = 0.f;
        for (int j = 0; j < HID; ++j) d += ctxv[j] * rgate_w[j];
        ctxdot = d;
    }
    __syncthreads();

    // Read gate per slot.
    if (tid < MSLOTS) {
        unsigned k = selkey[tid];
        int tok = token_of_rank[k >> 11];
        float lg = ctxdot + premem[tok] + rgate_b[0];
        float p = 1.f / (1.f + expf(-lg));
        rp[tid] = p;
        seltok[tid] = tok;
        read_prob_out[(size_t)b * MSLOTS + tid] = p;
    }
    __syncthreads();

    // Weighted retrieval.
    if (tid < HID) {
        float den = 1e-8f;
        #pragma unroll
        for (int m = 0; m < MSLOTS; ++m) den += rp[m];
        float acc = 0.f;
        #pragma unroll
        for (int m = 0; m < MSLOTS; ++m) acc += rp[m] * embed[seltok[m] * HID + tid];
        retrieved_out[(size_t)b * HID + tid] = acc / den;
    }
}

// ---------------------------------------------------------------------------
// Kernel 3: logits = retrieved(2048x64) @ head_w(64x128) + head_b
// fp32 WMMA: V_WMMA_F32_16X16X4_F32, one wave per 16x16 tile, 16 K-steps of 4.
// A (16x4 f32) layout: lanes 0-15 -> M=lane, VGPR{0,1}=K{0,1};
//                      lanes 16-31 -> M=lane-16, VGPR{0,1}=K{2,3}.
// B (4x16 f32) layout (dual): lanes 0-15 -> N=lane, K{0,1}; lanes 16-31 K{2,3}.
// C/D: lane<16 -> N=lane, M=v; lane>=16 -> N=lane-16, M=v+8.
// ---------------------------------------------------------------------------
__global__ void __launch_bounds__(256)
head_gemm_kernel(const float* __restrict__ A,     // (2048,64) retrieved
                 const float* __restrict__ W,     // (64,128) head_w
                 const float* __restrict__ bias,  // (128,)
                 float* __restrict__ out) {       // (2048,128)
    const int wave = threadIdx.x >> 5;           // 8 waves / block
    const int lane = threadIdx.x & 31;
    const int tile = blockIdx.x * 8 + wave;      // 1024 tiles total
    const int tm = tile >> 3;                    // 128 row tiles
    const int tn = tile & 7;                     // 8 col tiles
    const int rowBase = tm * 16;
    const int colBase = tn * 16;
    const int lhalf = lane >> 4;
    const int lmod = lane & 15;

    const float* arow = A + (size_t)(rowBase + lmod) * HID + (lhalf ? 2 : 0);
    const float* bcol = W + colBase + lmod + (lhalf ? 2 * VOCAB : 0);

    v8f c = {};
    #pragma unroll
    for (int kk = 0; kk < 16; ++kk) {
        const int k0 = kk * 4;
        v2f a, bb;
        a.x = arow[k0];
        a.y = arow[k0 + 1];
        bb.x = bcol[k0 * VOCAB];
        bb.y = bcol[k0 * VOCAB + VOCAB];
        c = __builtin_amdgcn_wmma_f32_16x16x4_f32(
                /*neg_a=*/false, a, /*neg_b=*/false, bb,
                /*c_mod=*/(short)0, c, /*reuse_a=*/false, /*reuse_b=*/false);
    }

    const float bj = bias[colBase + lmod];
    #pragma unroll
    for (int v = 0; v < 8; ++v) {
        const int M = v + lhalf * 8;
        out[(size_t)(rowBase + M) * VOCAB + colBase + lmod] = c[v] + bj;
    }
}

// ---------------------------------------------------------------------------
extern "C" void kernel_launch(void* const* d_in, const int* in_sizes, int n_in,
                              void* d_out, int out_size, void* d_ws, size_t ws_size,
                              hipStream_t stream) {
    const int*   seq     = (const int*)  d_in[0];
    const float* embed   = (const float*)d_in[1];
    const float* wgate_w = (const float*)d_in[2];
    const float* wgate_b = (const float*)d_in[3];
    const float* rgate_w = (const float*)d_in[4];
    const float* rgate_b = (const float*)d_in[5];
    const float* head_w  = (const float*)d_in[6];
    const float* head_b  = (const float*)d_in[7];

    float* out = (float*)d_out;
    float* logits_out     = out;                                   // (B,V)
    float* write_prob_out = out + (size_t)B_ROWS * VOCAB;          // (B,S)
    float* read_prob_out  = write_prob_out + (size_t)B_ROWS * S_LEN; // (B,M)

    // Workspace layout (bytes): token_sig[128]f | premem[128]f | rank_of[128]i
    //                           | token_of_rank[128]i | retrieved[B*H]f
    char* ws = (char*)d_ws;
    float* token_sig     = (float*)(ws + 0);
    float* premem        = (float*)(ws + 512);
    int*   rank_of       = (int*)  (ws + 1024);
    int*   token_of_rank = (int*)  (ws + 1536);
    float* retrieved     = (float*)(ws + 2048);

    token_tables_kernel<<<1, VOCAB, 0, stream>>>(
        embed, wgate_w, wgate_b, rgate_w, token_sig, premem, rank_of, token_of_rank);

    row_kernel<<<B_ROWS, 256, 0, stream>>>(
        seq, embed, rgate_w, rgate_b, token_sig, premem, rank_of, token_of_rank,
        write_prob_out, read_prob_out, retrieved);

    head_gemm_kernel<<<(B_ROWS / 16) * (VOCAB / 16) / 8, 256, 0, stream>>>(
        retrieved, head_w, head_b, logits_out);
}